// EventMamba_11055245820010
// MI455X (gfx1250) — compile-verified
//
#include <hip/hip_runtime.h>
#include <math.h>

// ---------------------------------------------------------------------------
// EventMamba forward for MI455X (gfx1250, wave32, WMMA).
// All channel-mixing matmuls run through one fused WMMA f32 16x16x4 GEMM
// kernel (2x2 register tiling, software-pipelined K loop, fused epilogues).
// Sequential parts (FPS, SSM scan) parallelize over the 128 point clouds.
// ---------------------------------------------------------------------------

typedef __attribute__((ext_vector_type(2))) float v2f;
typedef __attribute__((ext_vector_type(8))) float v8f;

static constexpr float BN_INV_F = 0.99999500003749968f; // 1/sqrt(1+1e-5)

enum { EP_NONE = 0, EP_BNRELU = 1, EP_BNRELU_RES = 2, EP_ADDRES = 3, EP_GELU = 4, EP_SOFTPLUS = 5 };

#define WMMA_F32(A, B, Cc) __builtin_amdgcn_wmma_f32_16x16x4_f32(false, (A), false, (B), (short)0, (Cc), false, false)

// ---------------------------------------------------------------------------
// Fused GEMM:  Y[P][O] = epilogue( X[P][C] * W^T )   (W is [O][C] row-major)
// One wave computes a 32x32 tile = 2x2 WMMA accumulators; K in steps of 4.
// A lane map : lane L (m = L&15): a.x = X[m][k0+2*(L>>4)], a.y = next k
// B lane map : lane L (n = L&15): b.x = W[n][k0+2*(L>>4)], b.y = next k
// D lane map : vgpr v, lane L   : element (m = v + 8*(L>>4), n = L&15)
// Out-of-range rows/cols are clamped at the pointer level (loads stay legal
// and unconditional); stores are guarded.  Only a C%4 tail is predicated.
// ---------------------------------------------------------------------------
__global__ void __launch_bounds__(128)
gemm_wmma(const float* __restrict__ X, int ldx,
          const float* __restrict__ W,
          const float* __restrict__ bias,
          const float* __restrict__ gain,
          const float* __restrict__ shift,
          const float* __restrict__ res, int ldr,
          float* __restrict__ Y, int ldy,
          int P, int C, int O, int mode, float bnmul)
{
    const int wave = threadIdx.x >> 5;
    const int lane = threadIdx.x & 31;
    const int half = lane >> 4;
    const int r    = lane & 15;
    const int p0 = (blockIdx.x * 4 + wave) * 32;
    const int o0 = blockIdx.y * 32;
    if (p0 >= P) return;                       // wave-uniform exit (EXEC stays full)

    const int pr0 = p0 + r,       pr1 = p0 + 16 + r;
    const int or0 = o0 + r,       or1 = o0 + 16 + r;
    const float* xr0 = X + (size_t)(pr0 < P ? pr0 : (P - 1)) * (size_t)ldx;
    const float* xr1 = X + (size_t)(pr1 < P ? pr1 : (P - 1)) * (size_t)ldx;
    const float* wr0 = W + (size_t)(or0 < O ? or0 : (O - 1)) * (size_t)C;
    const float* wr1 = W + (size_t)(or1 < O ? or1 : (O - 1)) * (size_t)C;

    v8f acc00 = {}, acc01 = {}, acc10 = {}, acc11 = {};
    const int kb = half << 1;                  // lane K phase: 0 or 2
    const int Cmain = C & ~3;

    if (Cmain > 0) {
        v2f a0, a1, b0, b1;
        a0.x = xr0[kb]; a0.y = xr0[kb + 1];
        a1.x = xr1[kb]; a1.y = xr1[kb + 1];
        b0.x = wr0[kb]; b0.y = wr0[kb + 1];
        b1.x = wr1[kb]; b1.y = wr1[kb + 1];
        for (int k0 = 4; k0 < Cmain; k0 += 4) {
            const int ka = k0 + kb;
            v2f na0, na1, nb0, nb1;
            na0.x = xr0[ka]; na0.y = xr0[ka + 1];
            na1.x = xr1[ka]; na1.y = xr1[ka + 1];
            nb0.x = wr0[ka]; nb0.y = wr0[ka + 1];
            nb1.x = wr1[ka]; nb1.y = wr1[ka + 1];
            acc00 = WMMA_F32(a0, b0, acc00);
            acc01 = WMMA_F32(a0, b1, acc01);
            acc10 = WMMA_F32(a1, b0, acc10);
            acc11 = WMMA_F32(a1, b1, acc11);
            a0 = na0; a1 = na1; b0 = nb0; b1 = nb1;
        }
        acc00 = WMMA_F32(a0, b0, acc00);
        acc01 = WMMA_F32(a0, b1, acc01);
        acc10 = WMMA_F32(a1, b0, acc10);
        acc11 = WMMA_F32(a1, b1, acc11);
    }
    if (Cmain < C) {                            // single guarded K tail
        const int ka = Cmain + kb;
        const bool e0 = ka < C, e1 = (ka + 1) < C;
        v2f a0, a1, b0, b1;
        a0.x = e0 ? xr0[ka] : 0.0f;  a0.y = e1 ? xr0[ka + 1] : 0.0f;
        a1.x = e0 ? xr1[ka] : 0.0f;  a1.y = e1 ? xr1[ka + 1] : 0.0f;
        b0.x = e0 ? wr0[ka] : 0.0f;  b0.y = e1 ? wr0[ka + 1] : 0.0f;
        b1.x = e0 ? wr1[ka] : 0.0f;  b1.y = e1 ? wr1[ka + 1] : 0.0f;
        acc00 = WMMA_F32(a0, b0, acc00);
        acc01 = WMMA_F32(a0, b1, acc01);
        acc10 = WMMA_F32(a1, b0, acc10);
        acc11 = WMMA_F32(a1, b1, acc11);
    }

    auto store_tile = [&](const v8f& acc, int pbase, int obase) {
        const int ocol = obase + r;
        if (ocol >= O) return;
        const float bb = bias  ? bias[ocol]         : 0.0f;
        const float gg = gain  ? gain[ocol] * bnmul : 1.0f;
        const float sh = shift ? shift[ocol]        : 0.0f;
        #pragma unroll
        for (int v = 0; v < 8; ++v) {
            const int mrow = pbase + v + (half << 3);
            if (mrow >= P) continue;
            float s = acc[v] + bb;
            switch (mode) {
            case EP_BNRELU:     s = fmaxf(s * gg + sh, 0.0f); break;
            case EP_BNRELU_RES: s = fmaxf(s * gg + sh + res[(size_t)mrow * ldr + ocol], 0.0f); break;
            case EP_ADDRES:     s = s + res[(size_t)mrow * ldr + ocol]; break;
            case EP_GELU:       s = 0.5f * s * (1.0f + erff(s * 0.70710678118654752f)); break;
            case EP_SOFTPLUS:   s = (s > 20.0f) ? s : log1pf(expf(s)); break;
            default: break;
            }
            Y[(size_t)mrow * ldy + ocol] = s;
        }
    };
    store_tile(acc00, p0,      o0);
    store_tile(acc01, p0,      o0 + 16);
    store_tile(acc10, p0 + 16, o0);
    store_tile(acc11, p0 + 16, o0 + 16);
}

// ---------------------------------------------------------------------------
// x (8,16,3,1024) -> xyz (128,1024,3)
// ---------------------------------------------------------------------------
__global__ void make_xyz(const float* __restrict__ x, float* __restrict__ xyz,
                         int M, int Np)
{
    const size_t total = (size_t)M * Np * 3;
    const size_t stride = (size_t)gridDim.x * blockDim.x;
    for (size_t g = (size_t)blockIdx.x * blockDim.x + threadIdx.x; g < total; g += stride) {
        const int m = (int)(g / ((size_t)Np * 3));
        const int rem = (int)(g - (size_t)m * Np * 3);
        const int n = rem / 3, c = rem - n * 3;
        xyz[g] = x[((size_t)m * 3 + c) * Np + n];
    }
}

// ---------------------------------------------------------------------------
// Farthest point sampling: one block per cloud, thread per point.
// ---------------------------------------------------------------------------
__global__ void fps_kernel(const float* __restrict__ xyz, int N, int S, int* __restrict__ fidx)
{
    const int m = blockIdx.x;
    const int t = threadIdx.x;
    extern __shared__ float smem[];
    float* sd = smem;
    int*   si = (int*)(smem + N);
    __shared__ int far;
    if (t == 0) far = 0;
    const float* base = xyz + (size_t)m * N * 3;
    const float px = base[t * 3], py = base[t * 3 + 1], pz = base[t * 3 + 2];
    float dist = 1e10f;
    __syncthreads();
    for (int it = 0; it < S; ++it) {
        const int f = far;
        if (t == 0) fidx[m * S + it] = f;
        const float cx = base[f * 3], cy = base[f * 3 + 1], cz = base[f * 3 + 2];
        const float dx = px - cx, dy = py - cy, dz = pz - cz;
        dist = fminf(dist, dx * dx + dy * dy + dz * dz);
        sd[t] = dist; si[t] = t;
        __syncthreads();
        for (int s = N >> 1; s > 0; s >>= 1) {
            if (t < s) {
                const float dv = sd[t + s]; const int iv = si[t + s];
                if (dv > sd[t] || (dv == sd[t] && iv < si[t])) { sd[t] = dv; si[t] = iv; }
            }
            __syncthreads();
        }
        if (t == 0) far = si[0];
        __syncthreads();
    }
}

// ---------------------------------------------------------------------------
// kNN (k=24) by insertion sort (stable -> matches top_k tie-breaking).
// ---------------------------------------------------------------------------
__global__ void __launch_bounds__(128)
knn_kernel(const float* __restrict__ xyz, const int* __restrict__ fidx,
           int N, int S, int* __restrict__ nidx, int M)
{
    const int gid = blockIdx.x * blockDim.x + threadIdx.x;
    if (gid >= M * S) return;
    const int m = gid / S;
    const float* base = xyz + (size_t)m * N * 3;
    const int ci = fidx[gid];
    const float cx = base[ci * 3], cy = base[ci * 3 + 1], cz = base[ci * 3 + 2];
    constexpr int K = 24;
    float bd[K]; int bi[K];
    #pragma unroll
    for (int k = 0; k < K; ++k) { bd[k] = 3.0e38f; bi[k] = 0; }
    for (int j = 0; j < N; ++j) {
        const float dx = base[j * 3] - cx, dy = base[j * 3 + 1] - cy, dz = base[j * 3 + 2] - cz;
        const float d = dx * dx + dy * dy + dz * dz;
        if (d < bd[K - 1]) {
            int p = K - 1;
            while (p > 0 && d < bd[p - 1]) { bd[p] = bd[p - 1]; bi[p] = bi[p - 1]; --p; }
            bd[p] = d; bi[p] = j;
        }
    }
    int* o = nidx + (size_t)gid * K;
    for (int k = 0; k < K; ++k) o[k] = bi[k];
}

__global__ void gather_xyz(const float* __restrict__ xyz, const int* __restrict__ fidx,
                           int N, int S, float* __restrict__ out, int M)
{
    const size_t total = (size_t)M * S * 3;
    const size_t stride = (size_t)gridDim.x * blockDim.x;
    for (size_t g = (size_t)blockIdx.x * blockDim.x + threadIdx.x; g < total; g += stride) {
        const int m = (int)(g / ((size_t)S * 3));
        const int rem = (int)(g - (size_t)m * S * 3);
        const int s = rem / 3, c = rem - s * 3;
        out[g] = xyz[((size_t)m * N + fidx[m * S + s]) * 3 + c];
    }
}

// Build point-major grouped tensor: row (m,s,k) holds [raw grouped C | mean C].
__global__ void gather_group(const float* __restrict__ pts, const int* __restrict__ nidx,
                             const int* __restrict__ fidx, int N, int S, int C,
                             float* __restrict__ out)
{
    const int g = blockIdx.x;                 // m*S + s
    const int m = g / S;
    constexpr int K = 24;
    const float* pbase = pts + (size_t)m * N * C;
    const float* ctr = pbase + (size_t)fidx[g] * C;
    for (int e = threadIdx.x; e < K * C; e += blockDim.x) {
        const int k = e / C, c = e - k * C;
        const int pi = nidx[(size_t)g * K + k];
        const size_t row = ((size_t)g * K + k) * (size_t)(2 * C);
        out[row + c] = pbase[(size_t)pi * C + c];
        out[row + C + c] = ctr[c];
    }
}

// Per-cloud std of (grouped - mean), ddof=1 (deterministic block reduction).
__global__ void group_std(const float* __restrict__ gbuf, int S, int C, float* __restrict__ stat)
{
    const int m = blockIdx.x;
    constexpr int K = 24;
    const size_t per = (size_t)S * K * C;
    const float* base = gbuf + (size_t)m * S * K * (size_t)(2 * C);
    float s1 = 0.0f, s2 = 0.0f;
    for (size_t e = threadIdx.x; e < per; e += blockDim.x) {
        const size_t rk = e / C; const int c = (int)(e - rk * C);
        const size_t row = rk * (size_t)(2 * C);
        const float d = base[row + c] - base[row + C + c];
        s1 += d; s2 += d * d;
    }
    __shared__ float a1[256], a2[256];
    a1[threadIdx.x] = s1; a2[threadIdx.x] = s2;
    __syncthreads();
    for (int s = 128; s > 0; s >>= 1) {
        if ((int)threadIdx.x < s) { a1[threadIdx.x] += a1[threadIdx.x + s]; a2[threadIdx.x] += a2[threadIdx.x + s]; }
        __syncthreads();
    }
    if (threadIdx.x == 0) {
        const float n = (float)per;
        const float mu = a1[0] / n;
        const float var = (a2[0] - n * mu * mu) / (n - 1.0f);
        stat[m] = sqrtf(fmaxf(var, 0.0f));
    }
}

__global__ void group_norm_apply(float* __restrict__ gbuf, const float* __restrict__ alpha,
                                 const float* __restrict__ beta, const float* __restrict__ stat,
                                 int S, int C, int M)
{
    constexpr int K = 24;
    const size_t total = (size_t)M * S * K * C;
    const size_t stride = (size_t)gridDim.x * blockDim.x;
    for (size_t g = (size_t)blockIdx.x * blockDim.x + threadIdx.x; g < total; g += stride) {
        const size_t rk = g / C; const int c = (int)(g - rk * C);
        const int m = (int)(rk / ((size_t)S * K));
        const size_t row = rk * (size_t)(2 * C);
        const float mean = gbuf[row + C + c];
        const float v = gbuf[row + c];
        gbuf[row + c] = alpha[c] * ((v - mean) / (stat[m] + 1e-5f)) + beta[c];
    }
}

// ---------------------------------------------------------------------------
// Attention pooling over G contiguous rows of C channels (one wave / group).
// ---------------------------------------------------------------------------
__global__ void __launch_bounds__(32)
att_pool(const float* __restrict__ X, int G, int C,
         const float* __restrict__ w, const float* __restrict__ b,
         float* __restrict__ out)
{
    const int grp = blockIdx.x;
    const int lane = threadIdx.x;
    const float* base = X + (size_t)grp * G * C;
    extern __shared__ float sc[];
    for (int g = 0; g < G; ++g) {
        float p = 0.0f;
        for (int c = lane; c < C; c += 32) p += base[(size_t)g * C + c] * w[c];
        for (int o = 16; o > 0; o >>= 1) p += __shfl_xor(p, o, 32);
        if (lane == 0) sc[g] = p + b[0];
    }
    __syncthreads();
    if (lane == 0) {
        float mx = -3.4e38f;
        for (int g = 0; g < G; ++g) mx = fmaxf(mx, sc[g]);
        float sm = 0.0f;
        for (int g = 0; g < G; ++g) { const float e = expf(sc[g] - mx); sc[g] = e; sm += e; }
        const float inv = 1.0f / sm;
        for (int g = 0; g < G; ++g) sc[g] *= inv;
    }
    __syncthreads();
    float* orow = out + (size_t)grp * C;
    for (int c = lane; c < C; c += 32) {
        float acc = 0.0f;
        for (int g = 0; g < G; ++g) acc += sc[g] * base[(size_t)g * C + c];
        orow[c] = acc;
    }
}

// ---------------------------------------------------------------------------
// Mamba pieces
// ---------------------------------------------------------------------------
__global__ void conv_silu(const float* __restrict__ xin, int ldx,
                          const float* __restrict__ w, const float* __restrict__ b,
                          int S, int di, int rev, float* __restrict__ out, int M)
{
    const size_t total = (size_t)M * S * di;
    const size_t stride = (size_t)gridDim.x * blockDim.x;
    for (size_t g = (size_t)blockIdx.x * blockDim.x + threadIdx.x; g < total; g += stride) {
        const size_t tok = g / di; const int ch = (int)(g - tok * di);
        const int m = (int)(tok / S); const int s = (int)(tok - (size_t)m * S);
        float acc = b[ch];
        #pragma unroll
        for (int j = 0; j < 4; ++j) {
            const int sp = rev ? (s + 3 - j) : (s - 3 + j);
            if (sp >= 0 && sp < S) acc += w[ch * 4 + j] * xin[((size_t)m * S + sp) * ldx + ch];
        }
        out[g] = acc / (1.0f + expf(-acc));     // silu
    }
}

__global__ void aneg_kernel(const float* __restrict__ al, float* __restrict__ an, int n)
{
    const int i = blockIdx.x * blockDim.x + threadIdx.x;
    if (i < n) an[i] = -expf(al[i]);
}

// Sequential selective scan: block per cloud, thread per channel (di).
__global__ void ssm_scan(const float* __restrict__ xc, const float* __restrict__ dt,
                         const float* __restrict__ dbl, int r,
                         const float* __restrict__ Aneg, int S, int di, int rev,
                         float* __restrict__ ys)
{
    const int m = blockIdx.x;
    const int ch = threadIdx.x;
    constexpr int NS = 16;
    float h[NS], a[NS];
    #pragma unroll
    for (int j = 0; j < NS; ++j) { h[j] = 0.0f; a[j] = Aneg[ch * NS + j]; }
    __shared__ float BC[32];
    const int r2n = r + 32;
    for (int t = 0; t < S; ++t) {
        const int s = rev ? (S - 1 - t) : t;
        const size_t tok = (size_t)m * S + s;
        if (threadIdx.x < 32) BC[threadIdx.x] = dbl[tok * r2n + r + threadIdx.x];
        __syncthreads();
        const float d  = dt[tok * di + ch];
        const float xv = xc[tok * di + ch];
        float acc = 0.0f;
        #pragma unroll
        for (int j = 0; j < NS; ++j) {
            h[j] = h[j] * expf(d * a[j]) + d * BC[j] * xv;
            acc += h[j] * BC[16 + j];
        }
        ys[tok * di + ch] = acc;
        __syncthreads();
    }
}

// out = (ys + xc*D) * silu(z)   (accumulate for backward branch)
__global__ void ssm_gate(const float* __restrict__ ys, const float* __restrict__ xc,
                         const float* __restrict__ D, const float* __restrict__ zsrc, int ldz,
                         float* __restrict__ out, size_t total, int di, int accum)
{
    const size_t stride = (size_t)gridDim.x * blockDim.x;
    for (size_t g = (size_t)blockIdx.x * blockDim.x + threadIdx.x; g < total; g += stride) {
        const size_t tok = g / di; const int ch = (int)(g - tok * di);
        const float z = zsrc[tok * ldz + ch];
        const float v = (ys[g] + xc[g] * D[ch]) * (z / (1.0f + expf(-z)));
        out[g] = accum ? (out[g] + v) : v;
    }
}

// ---------------------------------------------------------------------------
// LayerNorm (wave per token)
// ---------------------------------------------------------------------------
__global__ void __launch_bounds__(32)
layernorm(const float* __restrict__ x, const float* __restrict__ g,
          const float* __restrict__ b, float* __restrict__ y, int C)
{
    const int tok = blockIdx.x;
    const int lane = threadIdx.x;
    const float* row = x + (size_t)tok * C;
    float s = 0.0f;
    for (int c = lane; c < C; c += 32) s += row[c];
    for (int o = 16; o > 0; o >>= 1) s += __shfl_xor(s, o, 32);
    const float mean = s / C;
    float v = 0.0f;
    for (int c = lane; c < C; c += 32) { const float d = row[c] - mean; v += d * d; }
    for (int o = 16; o > 0; o >>= 1) v += __shfl_xor(v, o, 32);
    const float inv = rsqrtf(v / C + 1e-5f);
    float* yr = y + (size_t)tok * C;
    for (int c = lane; c < C; c += 32) yr[c] = (row[c] - mean) * inv * g[c] + b[c];
}

// ---------------------------------------------------------------------------
// One attention head (T=16, HD=64) with ALiBi bias, WMMA for QK^T and PV.
// ---------------------------------------------------------------------------
__global__ void __launch_bounds__(32)
attn_head(const float* __restrict__ q, const float* __restrict__ k,
          const float* __restrict__ v, float* __restrict__ o,
          int HD, int ldq, int NH)
{
    const int b = blockIdx.x / NH, hh = blockIdx.x % NH;
    const size_t base = (size_t)b * 16 * ldq + (size_t)hh * HD;
    const int lane = threadIdx.x, half = lane >> 4, r = lane & 15;

    v8f acc = {};
    for (int k0 = 0; k0 < HD; k0 += 4) {
        const int ka = k0 + (half << 1);
        v2f av, bv;
        av.x = q[base + (size_t)r * ldq + ka];
        av.y = q[base + (size_t)r * ldq + ka + 1];
        bv.x = k[base + (size_t)r * ldq + ka];
        bv.y = k[base + (size_t)r * ldq + ka + 1];
        acc = WMMA_F32(av, bv, acc);
    }

    __shared__ float sc[16][17];
    const float slope = (hh & 1) ? (1.0f / 24.0f) : 0.20412414523193150f; // 24^-1, 24^-0.5
    #pragma unroll
    for (int vv = 0; vv < 8; ++vv) {
        const int i = vv + (half << 3);
        const int j = r;
        float bias;
        if (hh < 2) bias = (j <= i) ? slope * (float)(j - i) : -1e9f;
        else        bias = (j >= i) ? slope * (float)(i - j) : -1e9f;
        sc[i][j] = acc[vv] * 0.125f + bias;
    }
    __syncthreads();
    if (lane < 16) {
        float mx = -3.4e38f;
        for (int j = 0; j < 16; ++j) mx = fmaxf(mx, sc[lane][j]);
        float sm = 0.0f;
        for (int j = 0; j < 16; ++j) { const float e = expf(sc[lane][j] - mx); sc[lane][j] = e; sm += e; }
        const float inv = 1.0f / sm;
        for (int j = 0; j < 16; ++j) sc[lane][j] *= inv;
    }
    __syncthreads();
    for (int ct = 0; ct < HD / 16; ++ct) {
        v8f po = {};
        #pragma unroll
        for (int k0 = 0; k0 < 16; k0 += 4) {
            const int ka = k0 + (half << 1);
            v2f av, bv;
            av.x = sc[r][ka];
            av.y = sc[r][ka + 1];
            bv.x = v[base + (size_t)ka * ldq + ct * 16 + r];
            bv.y = v[base + (size_t)(ka + 1) * ldq + ct * 16 + r];
            po = WMMA_F32(av, bv, po);
        }
        #pragma unroll
        for (int vv = 0; vv < 8; ++vv) {
            const int i = vv + (half << 3);
            o[base + (size_t)i * ldq + ct * 16 + r] = po[vv];
        }
    }
}

// ---------------------------------------------------------------------------
// Host-side wiring
// ---------------------------------------------------------------------------
struct Lin2P   { const float *W1,*b1,*g1,*bb1,*W2,*b2,*g2,*bb2; };
struct BranchP { const float *conv_w,*conv_b,*x_proj,*dt_w,*dt_b,*A_log,*D; };
struct MambaP  { const float *in_proj,*out_proj; BranchP f,b; };
struct TxP     { const float *Wq,*bq,*Wk,*bk,*Wv,*bv,*Wo,*bo,*D1,*d1,*D2,*d2,*ln1g,*ln1b,*ln2g,*ln2b; };

static inline unsigned blocks_for(size_t total)
{
    size_t b = (total + 255) / 256;
    if (b > 1048576) b = 1048576;
    return (unsigned)b;
}

extern "C" void kernel_launch(void* const* d_in, const int* in_sizes, int n_in,
                              void* d_out, int out_size, void* d_ws, size_t ws_size,
                              hipStream_t stream)
{
    (void)in_sizes; (void)n_in; (void)out_size; (void)ws_size;

    // ---- walk inputs in setup_inputs() insertion order ----
    int ii = 0;
    auto nxt = [&]() { return (const float*)d_in[ii++]; };
    const float* x = nxt();                                       // (8,16,3,1024)
    const float *g0_a = nxt(), *g0_b = nxt(), *g1_a = nxt(), *g1_b = nxt(), *g2_a = nxt(), *g2_b = nxt();
    const float *emb_W = nxt(), *emb_b = nxt(), *emb_g = nxt(), *emb_bb = nxt();
    auto lin2 = [&]() { Lin2P p; p.W1 = nxt(); p.b1 = nxt(); p.g1 = nxt(); p.bb1 = nxt();
                        p.W2 = nxt(); p.b2 = nxt(); p.g2 = nxt(); p.bb2 = nxt(); return p; };
    Lin2P conv1 = lin2(), conv1_1 = lin2(), conv2 = lin2(), conv2_1 = lin2(), conv3 = lin2(), conv3_1 = lin2();
    auto branch = [&]() { BranchP b; b.conv_w = nxt(); b.conv_b = nxt(); b.x_proj = nxt();
                          b.dt_w = nxt(); b.dt_b = nxt(); b.A_log = nxt(); b.D = nxt(); return b; };
    auto mambap = [&]() { MambaP m; m.in_proj = nxt(); m.out_proj = nxt(); m.f = branch(); m.b = branch(); return m; };
    MambaP mam1 = mambap(), mam2 = mambap(), mam3 = mambap();
    const float *att1_w = nxt(), *att1_b = nxt(), *att2_w = nxt(), *att2_b = nxt();
    const float *att3_w = nxt(), *att3_b = nxt(), *att4_w = nxt(), *att4_b = nxt();
    const float *cls_W1 = nxt(), *cls_b1 = nxt(), *cls_g = nxt(), *cls_bb = nxt();
    const float *cls_W2 = nxt(), *cls_b2 = nxt();
    TxP tx[6];
    for (int l = 0; l < 6; ++l) {
        TxP& t = tx[l];
        t.Wq = nxt(); t.bq = nxt(); t.Wk = nxt(); t.bk = nxt(); t.Wv = nxt(); t.bv = nxt();
        t.Wo = nxt(); t.bo = nxt(); t.D1 = nxt(); t.d1 = nxt(); t.D2 = nxt(); t.d2 = nxt();
        t.ln1g = nxt(); t.ln1b = nxt(); t.ln2g = nxt(); t.ln2b = nxt();
    }
    const float *lnout_g = nxt(), *lnout_b = nxt();

    // ---- workspace bump allocator ----
    char* ws = (char*)d_ws;
    size_t off = 0;
    auto allocf = [&](size_t n) -> float* {
        off = (off + 255) & ~size_t(255);
        float* p = (float*)(ws + off); off += n * sizeof(float); return p;
    };
    auto alloci = [&](size_t n) -> int* {
        off = (off + 255) & ~size_t(255);
        int* p = (int*)(ws + off); off += n * sizeof(int); return p;
    };

    const int M = 128, N0 = 1024, S1 = 512, S2 = 256, S3 = 128;
    constexpr int K = 24;

    float* xyz0 = allocf((size_t)M * N0 * 3);
    float* xyz1 = allocf((size_t)M * S1 * 3);
    float* xyz2 = allocf((size_t)M * S2 * 3);
    int*   fidx = alloci((size_t)M * S1);
    int*   nidx = alloci((size_t)M * S1 * K);
    float* stat = allocf(M);

    const size_t SLAB = (size_t)M * S1 * K * 64;       // 100.7M floats
    float* slabA = allocf(SLAB);
    float* slabB = allocf(SLAB);
    float* slabC = allocf(SLAB / 2);

    const size_t U = (size_t)M * S1 * 64;              // 4.19M floats (MS*c, stage-invariant)
    float* tok_in   = allocf(U);
    float* inproj   = allocf(4 * U);
    float* xcf      = allocf(2 * U);
    float* xcb      = allocf(2 * U);
    float* dblf     = allocf(U);
    float* dblb     = allocf(U);
    float* dtf      = allocf(2 * U);
    float* dtb      = allocf(2 * U);
    float* ysbuf    = allocf(2 * U);
    float* fbsum    = allocf(2 * U);
    float* tokout   = allocf(U);
    float* thid     = allocf(U);
    float* stageout = allocf(U);
    float* Aneg     = allocf(512 * 16);

    float* th  = allocf((size_t)128 * 256);
    float* thn = allocf((size_t)128 * 256);
    float* tq  = allocf((size_t)128 * 256);
    float* tk  = allocf((size_t)128 * 256);
    float* tv  = allocf((size_t)128 * 256);
    float* tao = allocf((size_t)128 * 256);
    float* tff = allocf((size_t)128 * 1024);
    float* tcl = allocf((size_t)128 * 512);

    auto gemm = [&](const float* X, int ldx, const float* W, const float* bias,
                    const float* gain, const float* shift, const float* res, int ldr,
                    float* Y, int ldy, int P, int C, int O, int mode, float bnmul) {
        dim3 grid((unsigned)((P + 127) / 128), (unsigned)((O + 31) / 32));
        gemm_wmma<<<grid, dim3(128), 0, stream>>>(X, ldx, W, bias, gain, shift, res, ldr,
                                                  Y, ldy, P, C, O, mode, bnmul);
    };

    auto run_mamba = [&](const MambaP& mp, int S, int c) {
        const int MS = M * S, di = 2 * c, r = c / 16, r2n = r + 32, di2 = 4 * c;
        gemm(tok_in, c, mp.in_proj, nullptr, nullptr, nullptr, nullptr, 0,
             inproj, di2, MS, c, di2, EP_NONE, 1.0f);
        const BranchP* brs[2] = { &mp.f, &mp.b };
        float* xcs[2]  = { xcf, xcb };
        float* dbls[2] = { dblf, dblb };
        float* dts[2]  = { dtf, dtb };
        for (int bi = 0; bi < 2; ++bi) {
            const BranchP& br = *brs[bi];
            float* xc = xcs[bi]; float* dbl = dbls[bi]; float* dt = dts[bi];
            const size_t tot = (size_t)MS * di;
            conv_silu<<<blocks_for(tot), 256, 0, stream>>>(inproj, di2, br.conv_w, br.conv_b,
                                                           S, di, bi, xc, M);
            gemm(xc, di, br.x_proj, nullptr, nullptr, nullptr, nullptr, 0,
                 dbl, r2n, MS, di, r2n, EP_NONE, 1.0f);
            gemm(dbl, r2n, br.dt_w, br.dt_b, nullptr, nullptr, nullptr, 0,
                 dt, di, MS, r, di, EP_SOFTPLUS, 1.0f);
            aneg_kernel<<<blocks_for((size_t)di * 16), 256, 0, stream>>>(br.A_log, Aneg, di * 16);
            ssm_scan<<<M, di, 0, stream>>>(xc, dt, dbl, r, Aneg, S, di, bi, ysbuf);
            ssm_gate<<<blocks_for(tot), 256, 0, stream>>>(ysbuf, xc, br.D, inproj + di, di2,
                                                          fbsum, tot, di, bi);
        }
        gemm(fbsum, di, mp.out_proj, nullptr, nullptr, nullptr, nullptr, 0,
             tokout, c, MS, di, c, EP_NONE, 1.0f);
    };

    // ================= Stage 1 =================
    make_xyz<<<blocks_for((size_t)M * N0 * 3), 256, 0, stream>>>(x, xyz0, M, N0);
    fps_kernel<<<M, N0, N0 * 8, stream>>>(xyz0, N0, S1, fidx);
    knn_kernel<<<(M * S1 + 127) / 128, 128, 0, stream>>>(xyz0, fidx, N0, S1, nidx, M);
    gather_xyz<<<blocks_for((size_t)M * S1 * 3), 256, 0, stream>>>(xyz0, fidx, N0, S1, xyz1, M);
    gather_group<<<M * S1, 128, 0, stream>>>(xyz0, nidx, fidx, N0, S1, 3, slabA);
    group_std<<<M, 256, 0, stream>>>(slabA, S1, 3, stat);
    group_norm_apply<<<blocks_for((size_t)M * S1 * K * 3), 256, 0, stream>>>(slabA, g0_a, g0_b, stat, S1, 3, M);
    const int P1 = M * S1 * K;
    gemm(slabA, 6, emb_W, emb_b, emb_g, emb_bb, nullptr, 0, slabB, 64, P1, 6, 64, EP_BNRELU, BN_INV_F);
    gemm(slabB, 64, conv1.W1, conv1.b1, conv1.g1, conv1.bb1, nullptr, 0, slabC, 32, P1, 64, 32, EP_BNRELU, BN_INV_F);
    gemm(slabC, 32, conv1.W2, conv1.b2, conv1.g2, conv1.bb2, slabB, 64, slabA, 64, P1, 32, 64, EP_BNRELU_RES, BN_INV_F);
    att_pool<<<M * S1, 32, K * sizeof(float), stream>>>(slabA, K, 64, att1_w, att1_b, tok_in);
    run_mamba(mam1, S1, 64);
    const int MS1 = M * S1;
    gemm(tokout, 64, conv1_1.W1, conv1_1.b1, conv1_1.g1, conv1_1.bb1, nullptr, 0, thid, 32, MS1, 64, 32, EP_BNRELU, BN_INV_F);
    gemm(thid, 32, conv1_1.W2, conv1_1.b2, conv1_1.g2, conv1_1.bb2, tokout, 64, stageout, 64, MS1, 32, 64, EP_BNRELU_RES, BN_INV_F);

    // ================= Stage 2 =================
    fps_kernel<<<M, S1, S1 * 8, stream>>>(xyz1, S1, S2, fidx);
    knn_kernel<<<(M * S2 + 127) / 128, 128, 0, stream>>>(xyz1, fidx, S1, S2, nidx, M);
    gather_xyz<<<blocks_for((size_t)M * S2 * 3), 256, 0, stream>>>(xyz1, fidx, S1, S2, xyz2, M);
    gather_group<<<M * S2, 128, 0, stream>>>(stageout, nidx, fidx, S1, S2, 64, slabA);
    group_std<<<M, 256, 0, stream>>>(slabA, S2, 64, stat);
    group_norm_apply<<<blocks_for((size_t)M * S2 * K * 64), 256, 0, stream>>>(slabA, g1_a, g1_b, stat, S2, 64, M);
    const int P2 = M * S2 * K;
    gemm(slabA, 128, conv2.W1, conv2.b1, conv2.g1, conv2.bb1, nullptr, 0, slabC, 64, P2, 128, 64, EP_BNRELU, BN_INV_F);
    gemm(slabC, 64, conv2.W2, conv2.b2, conv2.g2, conv2.bb2, slabA, 128, slabB, 128, P2, 64, 128, EP_BNRELU_RES, BN_INV_F);
    att_pool<<<M * S2, 32, K * sizeof(float), stream>>>(slabB, K, 128, att2_w, att2_b, tok_in);
    run_mamba(mam2, S2, 128);
    const int MS2 = M * S2;
    gemm(tokout, 128, conv2_1.W1, conv2_1.b1, conv2_1.g1, conv2_1.bb1, nullptr, 0, thid, 64, MS2, 128, 64, EP_BNRELU, BN_INV_F);
    gemm(thid, 64, conv2_1.W2, conv2_1.b2, conv2_1.g2, conv2_1.bb2, tokout, 128, stageout, 128, MS2, 64, 128, EP_BNRELU_RES, BN_INV_F);

    // ================= Stage 3 =================
    fps_kernel<<<M, S2, S2 * 8, stream>>>(xyz2, S2, S3, fidx);
    knn_kernel<<<(M * S3 + 127) / 128, 128, 0, stream>>>(xyz2, fidx, S2, S3, nidx, M);
    gather_group<<<M * S3, 128, 0, stream>>>(stageout, nidx, fidx, S2, S3, 128, slabA);
    group_std<<<M, 256, 0, stream>>>(slabA, S3, 128, stat);
    group_norm_apply<<<blocks_for((size_t)M * S3 * K * 128), 256, 0, stream>>>(slabA, g2_a, g2_b, stat, S3, 128, M);
    const int P3 = M * S3 * K;
    gemm(slabA, 256, conv3.W1, conv3.b1, conv3.g1, conv3.bb1, nullptr, 0, slabC, 128, P3, 256, 128, EP_BNRELU, BN_INV_F);
    gemm(slabC, 128, conv3.W2, conv3.b2, conv3.g2, conv3.bb2, slabA, 256, slabB, 256, P3, 128, 256, EP_BNRELU_RES, BN_INV_F);
    att_pool<<<M * S3, 32, K * sizeof(float), stream>>>(slabB, K, 256, att3_w, att3_b, tok_in);
    run_mamba(mam3, S3, 256);
    const int MS3 = M * S3;
    gemm(tokout, 256, conv3_1.W1, conv3_1.b1, conv3_1.g1, conv3_1.bb1, nullptr, 0, thid, 128, MS3, 256, 128, EP_BNRELU, BN_INV_F);
    gemm(thid, 128, conv3_1.W2, conv3_1.b2, conv3_1.g2, conv3_1.bb2, tokout, 256, stageout, 256, MS3, 128, 256, EP_BNRELU_RES, BN_INV_F);

    // ================= Global attention pool -> (8,16,256) tokens =========
    att_pool<<<M, 32, S3 * sizeof(float), stream>>>(stageout, S3, 256, att4_w, att4_b, th);

    // ================= Transformer (6 layers, T=16, B=8, 4 heads x 64) ====
    for (int l = 0; l < 6; ++l) {
        layernorm<<<128, 32, 0, stream>>>(th, tx[l].ln1g, tx[l].ln1b, thn, 256);
        gemm(thn, 256, tx[l].Wq, tx[l].bq, nullptr, nullptr, nullptr, 0, tq, 256, 128, 256, 256, EP_NONE, 1.0f);
        gemm(thn, 256, tx[l].Wk, tx[l].bk, nullptr, nullptr, nullptr, 0, tk, 256, 128, 256, 256, EP_NONE, 1.0f);
        gemm(thn, 256, tx[l].Wv, tx[l].bv, nullptr, nullptr, nullptr, 0, tv, 256, 128, 256, 256, EP_NONE, 1.0f);
        attn_head<<<8 * 4, 32, 0, stream>>>(tq, tk, tv, tao, 64, 256, 4);
        gemm(tao, 256, tx[l].Wo, tx[l].bo, nullptr, nullptr, th, 256, th, 256, 128, 256, 256, EP_ADDRES, 1.0f);
        layernorm<<<128, 32, 0, stream>>>(th, tx[l].ln2g, tx[l].ln2b, thn, 256);
        gemm(thn, 256, tx[l].D1, tx[l].d1, nullptr, nullptr, nullptr, 0, tff, 1024, 128, 256, 1024, EP_GELU, 1.0f);
        gemm(tff, 1024, tx[l].D2, tx[l].d2, nullptr, nullptr, th, 256, th, 256, 128, 1024, 256, EP_ADDRES, 1.0f);
    }

    // ================= Classifier ========================================
    layernorm<<<128, 32, 0, stream>>>(th, lnout_g, lnout_b, thn, 256);
    gemm(thn, 256, cls_W1, cls_b1, cls_g, cls_bb, nullptr, 0, tcl, 512, 128, 256, 512, EP_BNRELU, BN_INV_F);
    gemm(tcl, 512, cls_W2, cls_b2, nullptr, nullptr, nullptr, 0, (float*)d_out, 2, 128, 512, 2, EP_NONE, 1.0f);
}